// SurfaceDistanceMeanLoss_59511066853608
// MI455X (gfx1250) — compile-verified
//
#include <hip/hip_runtime.h>
#include <hip/hip_bf16.h>

// CDNA5 / gfx1250, wave32.
typedef float v2f __attribute__((ext_vector_type(2)));
typedef float v8f __attribute__((ext_vector_type(8)));

#define HH 512
#define WW 512
#define NPIX (HH * WW)
#define BIGF 1.0e6f

// ---------------------------------------------------------------------------
// Stage 1: surface masks S = p & ~erode(p) (cross structure, border=False),
// plus per-block partial sums of |S|+|S'| for the denominator.
// ---------------------------------------------------------------------------
__global__ void k_surface(const float* __restrict__ pred,
                          const float* __restrict__ gt,
                          float* __restrict__ maskS,
                          float* __restrict__ maskSp,
                          float* __restrict__ denPart) {
    __shared__ float red[256];
    const int idx = blockIdx.x * 256 + threadIdx.x;     // 0..NPIX-1
    const int r = idx >> 9;
    const int c = idx & 511;
    float dsum = 0.0f;
    {
        const bool p  = pred[idx] != 0.0f;
        const bool up = (r > 0)   && (pred[idx - WW] != 0.0f);
        const bool dn = (r < 511) && (pred[idx + WW] != 0.0f);
        const bool lf = (c > 0)   && (pred[idx - 1]  != 0.0f);
        const bool rt = (c < 511) && (pred[idx + 1]  != 0.0f);
        const bool s = p && !(up && dn && lf && rt);
        const float sf = s ? 1.0f : 0.0f;
        maskS[idx] = sf;
        dsum += sf;
    }
    {
        const bool p  = gt[idx] != 0.0f;
        const bool up = (r > 0)   && (gt[idx - WW] != 0.0f);
        const bool dn = (r < 511) && (gt[idx + WW] != 0.0f);
        const bool lf = (c > 0)   && (gt[idx - 1]  != 0.0f);
        const bool rt = (c < 511) && (gt[idx + 1]  != 0.0f);
        const bool s = p && !(up && dn && lf && rt);
        const float sf = s ? 1.0f : 0.0f;
        maskSp[idx] = sf;
        dsum += sf;
    }
    red[threadIdx.x] = dsum;
    __syncthreads();
    for (int off = 128; off > 0; off >>= 1) {
        if (threadIdx.x < off) red[threadIdx.x] += red[threadIdx.x + off];
        __syncthreads();
    }
    if (threadIdx.x == 0) denPart[blockIdx.x] = red[0];
}

// ---------------------------------------------------------------------------
// Stage 2: vertical EDT pass. One thread per column (coalesced across the
// row), two sweeps; leaves g^2 in the field buffer.
// gid < 512 -> field A (pred surface), gid >= 512 -> field B (gt surface).
// ---------------------------------------------------------------------------
__global__ void k_colsweep(const float* __restrict__ maskS,
                           const float* __restrict__ maskSp,
                           float* __restrict__ fieldA,
                           float* __restrict__ fieldB) {
    const int gid = blockIdx.x * 256 + threadIdx.x;     // 0..1023
    const float* __restrict__ m = (gid < 512) ? maskS : maskSp;
    float* __restrict__ out     = (gid < 512) ? fieldA : fieldB;
    const int c = gid & 511;
    float carry = BIGF;
    for (int r = 0; r < HH; ++r) {
        const float s = m[r * WW + c];
        carry = (s != 0.0f) ? 0.0f : (carry + 1.0f);
        out[r * WW + c] = carry;                        // down-sweep distance
    }
    carry = BIGF;
    for (int r = HH - 1; r >= 0; --r) {
        const float s = m[r * WW + c];
        carry = (s != 0.0f) ? 0.0f : (carry + 1.0f);
        const float g = fminf(out[r * WW + c], carry);
        out[r * WW + c] = g * g;                        // store g^2
    }
}

// ---------------------------------------------------------------------------
// Stage 3: row-wise exact min-plus: d2[c] = min_j (g2[j] + (c-j)^2).
// g2 row staged in LDS (2 KB of the 320 KB/WGP); result sqrt'ed in place.
// Tropical semiring -> VALU fma+min, cannot be a WMMA.
// ---------------------------------------------------------------------------
__global__ void k_rowpass(float* __restrict__ fieldA,
                          float* __restrict__ fieldB) {
    __shared__ float g2[WW];
    const int r = blockIdx.x;
    float* __restrict__ row = ((blockIdx.y == 0) ? fieldA : fieldB) + r * WW;
    g2[threadIdx.x]       = row[threadIdx.x];
    g2[threadIdx.x + 256] = row[threadIdx.x + 256];
    __syncthreads();
    const int c0 = threadIdx.x, c1 = threadIdx.x + 256;
    const float fc0 = (float)c0, fc1 = (float)c1;
    float m0 = 3.4e38f, m1 = 3.4e38f;
    float fj = 0.0f;
#pragma unroll 8
    for (int j = 0; j < WW; ++j) {
        const float v = g2[j];
        const float d0 = fc0 - fj;
        const float d1 = fc1 - fj;
        m0 = fminf(m0, fmaf(d0, d0, v));
        m1 = fminf(m1, fmaf(d1, d1, v));
        fj += 1.0f;
    }
    row[c0] = sqrtf(m0);
    row[c1] = sqrtf(m1);
}

// ---------------------------------------------------------------------------
// Stage 4: masked-sum via WMMA trace trick (exact FP32 matrix path).
//   sum_{r,c} dt[r,c]*mask[r,c] over a 16-row strip
//     = trace( A(16xK) * B(Kx16) ) with A = dt rows, B = mask^T.
// V_WMMA_F32_16X16X4_F32 accumulates D += A*B over K-chunks of 4.
// A layout (ISA 7.12.2, 32-bit 16x4): lane l -> M = l&15, K = 2*(l>>4)+{0,1}
//   -> contiguous float2 per lane. B (4x16) mirrors C's striping:
//   VGPR0: K=0|K=2, VGPR1: K=1|K=3 -> also a contiguous float2 per lane.
// One wave32 per block, no divergence (EXEC all ones). Prefetches are
// unconditional: worst-case targets stay inside the mapped workspace, so
// no guard -> no scalar branches in the hot loop.
// Diagonal of the 16x16 F32 accumulator: VGPR v -> lanes v and v+24.
// ---------------------------------------------------------------------------
__global__ void k_wmma_reduce(const float* __restrict__ fieldA,
                              const float* __restrict__ fieldB,
                              const float* __restrict__ maskS,
                              const float* __restrict__ maskSp,
                              float* __restrict__ numPart) {
    const int lane  = threadIdx.x;           // 0..31, wave32
    const int r0    = blockIdx.x * 16;       // 32 row strips
    const int cbase = blockIdx.y * 128;      // 4 column quarters
    const float* __restrict__ dt;
    const float* __restrict__ msk;
    if (blockIdx.z == 0) { dt = fieldA; msk = maskSp; }   // sum dta * S'
    else                 { dt = fieldB; msk = maskS;  }   // sum dtb * S
    const int mrow = lane & 15;
    const int koff = (lane >> 4) * 2;
    const float* aptr = dt  + (r0 + mrow) * WW + cbase + koff;  // A: M=mrow
    const float* bptr = msk + (r0 + mrow) * WW + cbase + koff;  // B: N=mrow
    v8f acc = {0.f, 0.f, 0.f, 0.f, 0.f, 0.f, 0.f, 0.f};
#pragma unroll 8
    for (int k = 0; k < 32; ++k) {           // 32 K-chunks of 4 -> 128 cols
        const v2f a = *(const v2f*)(aptr + k * 4);
        const v2f b = *(const v2f*)(bptr + k * 4);
        // gfx1250 global_prefetch_b8, unconditional (targets remain inside
        // the workspace allocation; speculative prefetch is drop-safe).
        __builtin_prefetch(aptr + (k + 8) * 4, 0, 1);
        __builtin_prefetch(bptr + (k + 8) * 4, 0, 1);
        acc = __builtin_amdgcn_wmma_f32_16x16x4_f32(
            /*neg_a=*/false, a, /*neg_b=*/false, b,
            /*c_mod=*/(short)0, acc, /*reuse_a=*/false, /*reuse_b=*/false);
    }
    // Extract trace: C VGPR v holds D[v][lane] (lanes 0-15) and
    // D[v+8][lane-16] (lanes 16-31); diagonal sits at lanes v and v+24.
    float tsum = 0.0f;
#pragma unroll
    for (int v = 0; v < 8; ++v) {
        const float val = acc[v];
        tsum += ((lane == v) || (lane == v + 24)) ? val : 0.0f;
    }
    for (int off = 16; off > 0; off >>= 1)
        tsum += __shfl_xor(tsum, off, 32);
    if (lane == 0)
        numPart[blockIdx.x + 32 * (blockIdx.y + 4 * blockIdx.z)] = tsum;
}

// ---------------------------------------------------------------------------
// Stage 5: deterministic finalize (fixed reduction tree, no atomics).
// ---------------------------------------------------------------------------
__global__ void k_finalize(const float* __restrict__ denPart,
                           const float* __restrict__ numPart,
                           float* __restrict__ out) {
    __shared__ float sden[256];
    __shared__ float snum[256];
    const int t = threadIdx.x;
    sden[t] = denPart[t] + denPart[t + 256] + denPart[t + 512] + denPart[t + 768];
    snum[t] = numPart[t];
    __syncthreads();
    for (int off = 128; off > 0; off >>= 1) {
        if (t < off) { sden[t] += sden[t + off]; snum[t] += snum[t + off]; }
        __syncthreads();
    }
    if (t == 0) out[0] = snum[0] / sden[0];
}

// ---------------------------------------------------------------------------
extern "C" void kernel_launch(void* const* d_in, const int* in_sizes, int n_in,
                              void* d_out, int out_size, void* d_ws, size_t ws_size,
                              hipStream_t stream) {
    (void)in_sizes; (void)n_in; (void)out_size; (void)ws_size;
    const float* pred = (const float*)d_in[0];
    const float* gt   = (const float*)d_in[1];
    float* ws      = (float*)d_ws;
    float* maskS   = ws;                 // [NPIX]
    float* maskSp  = ws + NPIX;          // [NPIX]
    float* fieldA  = ws + 2 * NPIX;      // [NPIX] down-temp -> g2 -> edt(S)
    float* fieldB  = ws + 3 * NPIX;      // [NPIX] edt(S')
    float* denPart = ws + 4 * NPIX;      // [1024]
    float* numPart = denPart + 1024;     // [256]
    float* out     = (float*)d_out;

    k_surface<<<dim3(NPIX / 256), dim3(256), 0, stream>>>(pred, gt, maskS, maskSp, denPart);
    k_colsweep<<<dim3(4), dim3(256), 0, stream>>>(maskS, maskSp, fieldA, fieldB);
    k_rowpass<<<dim3(HH, 2), dim3(256), 0, stream>>>(fieldA, fieldB);
    k_wmma_reduce<<<dim3(32, 4, 2), dim3(32), 0, stream>>>(fieldA, fieldB, maskS, maskSp, numPart);
    k_finalize<<<dim3(1), dim3(256), 0, stream>>>(denPart, numPart, out);
}